// LIFSpike_90074054132401
// MI455X (gfx1250) — compile-verified
//
#include <hip/hip_runtime.h>

// LIF forward scan, Vth = 1.0:
//   u_t = u_{t-1} - o_{t-1} + x_t ;  o_t = (u_t > 1.0f) ? 1.0f : 0.0f
// x, out: [b=32, c=128, h=32, w=32, t=16] float32, t innermost (stride 1).
// Each thread owns one (b,c,h,w) position: 16 contiguous floats in, 16 out.
// Memory-bound (512 MB streamed @ 23.3 TB/s ~ 22 us); all state in VGPRs,
// non-temporal hints so the single-use stream doesn't thrash the 192 MB L2.

#define T_STEPS 16
#define VTH 1.0f

// True clang vector type: __builtin_nontemporal_* requires a vector of
// scalars, not HIP's float4 class. Swizzles (.x/.y/.z/.w) still work.
typedef float v4f __attribute__((ext_vector_type(4)));

__global__ __launch_bounds__(256) void lif_scan_kernel(
    const float* __restrict__ x, float* __restrict__ out, int n_pos)
{
    const int idx = blockIdx.x * blockDim.x + threadIdx.x;
    if (idx >= n_pos) return;

    const v4f* __restrict__ xp = reinterpret_cast<const v4f*>(x)   + (size_t)idx * 4;
    v4f*       __restrict__ op = reinterpret_cast<v4f*>(out)       + (size_t)idx * 4;

    // 4x global_load_b128 th:TH_LOAD_NT — 64B per lane, coalesced across wave32.
    v4f v0 = __builtin_nontemporal_load(xp + 0);
    v4f v1 = __builtin_nontemporal_load(xp + 1);
    v4f v2 = __builtin_nontemporal_load(xp + 2);
    v4f v3 = __builtin_nontemporal_load(xp + 3);

    float u = 0.0f;
    float o = 0.0f;

    // Fully unrolled 16-step recurrence; overwrite the input registers with
    // the spike outputs in place to keep VGPR pressure minimal.
#define LIF_STEP(e)                          \
    {                                        \
        u = (u - o) + (e);                   \
        o = (u > VTH) ? 1.0f : 0.0f;         \
        (e) = o;                             \
    }

    LIF_STEP(v0.x) LIF_STEP(v0.y) LIF_STEP(v0.z) LIF_STEP(v0.w)
    LIF_STEP(v1.x) LIF_STEP(v1.y) LIF_STEP(v1.z) LIF_STEP(v1.w)
    LIF_STEP(v2.x) LIF_STEP(v2.y) LIF_STEP(v2.z) LIF_STEP(v2.w)
    LIF_STEP(v3.x) LIF_STEP(v3.y) LIF_STEP(v3.z) LIF_STEP(v3.w)

#undef LIF_STEP

    // 4x global_store_b128 th:TH_STORE_NT — spikes are written once.
    __builtin_nontemporal_store(v0, op + 0);
    __builtin_nontemporal_store(v1, op + 1);
    __builtin_nontemporal_store(v2, op + 2);
    __builtin_nontemporal_store(v3, op + 3);
}

extern "C" void kernel_launch(void* const* d_in, const int* in_sizes, int n_in,
                              void* d_out, int out_size, void* d_ws, size_t ws_size,
                              hipStream_t stream)
{
    const float* x   = (const float*)d_in[0];
    float*       out = (float*)d_out;

    // in_sizes[0] = b*c*h*w*t total elements; one thread per (b,c,h,w) position.
    const int n_pos = in_sizes[0] / T_STEPS;

    const int block = 256;                         // 8 wave32 waves per block
    const int grid  = (n_pos + block - 1) / block; // 16384 blocks for the ref shape

    lif_scan_kernel<<<grid, block, 0, stream>>>(x, out, n_pos);
}